// PostProcess_61564061221307
// MI455X (gfx1250) — compile-verified
//
#include <hip/hip_runtime.h>
#include <hip/hip_bf16.h>
#include <stdint.h>
#include <math.h>

#define QD 900
#define CD 91
#define NQC 81900      // 900*91 elements per batch row
#define NQC4 20475     // NQC/4 (exactly divisible; rows are 16B aligned)
#define NSEL 300
#define NTH 256
#define TILE 1024      // floats per async-staged tile (4KB)
#define FULL_TILES 79  // 79*1024 = 80896
#define TAIL_START 80896

// Order-preserving float->uint key: larger float <=> larger uint.
__device__ __forceinline__ uint32_t fkey(float x) {
  uint32_t u = __float_as_uint(x);
  return u ^ ((u & 0x80000000u) ? 0xFFFFFFFFu : 0x80000000u);
}
__device__ __forceinline__ float unfkey(uint32_t k) {
  uint32_t u = (k & 0x80000000u) ? (k ^ 0x80000000u) : ~k;
  return __uint_as_float(u);
}

// Find, scanning bins from high to low, the bin where the cumulative count
// reaches `need`. Writes threshold bin to *s_bin and remaining count within
// that bin to *s_rem. Parallel: chunk sums + Hillis-Steele inclusive scan.
__device__ __forceinline__ void find_bin(uint32_t* hist, uint32_t* scanbuf,
                                         int nbins, uint32_t need,
                                         uint32_t* s_bin, uint32_t* s_rem,
                                         int tid) {
  const int per = nbins / NTH;
  const int base = tid * per;
  uint32_t sum = 0;
  for (int j = 0; j < per; ++j) sum += hist[nbins - 1 - (base + j)];
  scanbuf[tid] = sum;
  __syncthreads();
  for (int off = 1; off < NTH; off <<= 1) {
    uint32_t v = (tid >= off) ? scanbuf[tid - off] : 0u;
    __syncthreads();
    scanbuf[tid] += v;
    __syncthreads();
  }
  uint32_t incl = scanbuf[tid];
  uint32_t excl = incl - sum;
  if (excl < need && incl >= need) {  // exactly one owner thread
    uint32_t cum = excl;
    for (int j = 0; j < per; ++j) {
      int bin = nbins - 1 - (base + j);
      uint32_t c = hist[bin];
      if (cum + c >= need) {
        *s_bin = (uint32_t)bin;
        *s_rem = need - cum;
        break;
      }
      cum += c;
    }
  }
  __syncthreads();
}

__global__ __launch_bounds__(NTH) void postprocess_topk_kernel(
    const float* __restrict__ logits, const float* __restrict__ boxes,
    const int* __restrict__ tsizes, float* __restrict__ out, int B) {
  __shared__ uint32_t hist[2048];
  __shared__ uint32_t scanbuf[NTH];
  __shared__ unsigned long long sel[1024];
  __shared__ float stage[2][TILE];
  __shared__ uint32_t s_fix, s_need, s_cnt;

  const int tid = threadIdx.x;
  const int b = blockIdx.x;
  const float* rowp = logits + (size_t)b * NQC;

  // ---------- Pass 1: histogram of key bits [31:21] ----------
  for (int i = tid; i < 2048; i += NTH) hist[i] = 0u;
  __syncthreads();

  // Double-buffered async global->LDS streaming (CDNA5 async-tensor path).
  // Each thread stages & consumes its own 16B, so only per-wave ASYNCcnt
  // waits are needed inside the pipeline (no workgroup barrier).
  uint32_t lds_off0 = (uint32_t)(uintptr_t)(void*)&stage[0][tid * 4];
  uint32_t lds_off1 = (uint32_t)(uintptr_t)(void*)&stage[1][tid * 4];
  {
    unsigned long long ga = (unsigned long long)(uintptr_t)(rowp + tid * 4);
    asm volatile("global_load_async_to_lds_b128 %0, %1, off"
                 :: "v"(lds_off0), "v"(ga) : "memory");
  }
  for (int t = 0; t < FULL_TILES; ++t) {
    const int buf = t & 1;
    if (t + 1 < FULL_TILES) {
      unsigned long long ga =
          (unsigned long long)(uintptr_t)(rowp + (t + 1) * TILE + tid * 4);
      asm volatile("global_load_async_to_lds_b128 %0, %1, off"
                   :: "v"(buf ? lds_off0 : lds_off1), "v"(ga) : "memory");
      asm volatile("s_wait_asynccnt 0x1" ::: "memory");  // tile t landed
    } else {
      asm volatile("s_wait_asynccnt 0x0" ::: "memory");
    }
    float4 v = *(const float4*)&stage[buf][tid * 4];
    atomicAdd(&hist[fkey(v.x) >> 21], 1u);
    atomicAdd(&hist[fkey(v.y) >> 21], 1u);
    atomicAdd(&hist[fkey(v.z) >> 21], 1u);
    atomicAdd(&hist[fkey(v.w) >> 21], 1u);
  }
  for (int i = TAIL_START + tid; i < NQC; i += NTH)  // 1004-elem tail
    atomicAdd(&hist[fkey(rowp[i]) >> 21], 1u);
  __syncthreads();

  find_bin(hist, scanbuf, 2048, (uint32_t)NSEL, &s_fix, &s_need, tid);
  const uint32_t bin1 = s_fix;
  const uint32_t rem1 = s_need;

  // ---------- Pass 2: bits [20:10] within bin1 (L2-resident rescan) ----------
  for (int i = tid; i < 2048; i += NTH) hist[i] = 0u;
  __syncthreads();
  const float4* rowp4 = (const float4*)rowp;
  for (int i = tid; i < NQC4; i += NTH) {
    float4 v = rowp4[i];
    uint32_t k;
    k = fkey(v.x); if ((k >> 21) == bin1) atomicAdd(&hist[(k >> 10) & 2047u], 1u);
    k = fkey(v.y); if ((k >> 21) == bin1) atomicAdd(&hist[(k >> 10) & 2047u], 1u);
    k = fkey(v.z); if ((k >> 21) == bin1) atomicAdd(&hist[(k >> 10) & 2047u], 1u);
    k = fkey(v.w); if ((k >> 21) == bin1) atomicAdd(&hist[(k >> 10) & 2047u], 1u);
  }
  __syncthreads();
  find_bin(hist, scanbuf, 2048, rem1, &s_fix, &s_need, tid);
  const uint32_t fix2 = (bin1 << 11) | s_fix;
  const uint32_t rem2 = s_need;

  // ---------- Pass 3: bits [9:0] within fix2 ----------
  for (int i = tid; i < 1024; i += NTH) hist[i] = 0u;
  __syncthreads();
  for (int i = tid; i < NQC4; i += NTH) {
    float4 v = rowp4[i];
    uint32_t k;
    k = fkey(v.x); if ((k >> 10) == fix2) atomicAdd(&hist[k & 1023u], 1u);
    k = fkey(v.y); if ((k >> 10) == fix2) atomicAdd(&hist[k & 1023u], 1u);
    k = fkey(v.z); if ((k >> 10) == fix2) atomicAdd(&hist[k & 1023u], 1u);
    k = fkey(v.w); if ((k >> 10) == fix2) atomicAdd(&hist[k & 1023u], 1u);
  }
  __syncthreads();
  find_bin(hist, scanbuf, 1024, rem2, &s_fix, &s_need, tid);
  const uint32_t T = (fix2 << 10) | s_fix;  // exact key of 300th largest

  // ---------- Compact all keys >= T into LDS ----------
  for (int i = tid; i < 1024; i += NTH) sel[i] = 0ull;  // pad = smallest
  if (tid == 0) s_cnt = 0u;
  __syncthreads();
  for (int i = tid; i < NQC4; i += NTH) {
    float4 v = rowp4[i];
    uint32_t ks[4] = {fkey(v.x), fkey(v.y), fkey(v.z), fkey(v.w)};
    const uint32_t bidx = (uint32_t)(i * 4);
#pragma unroll
    for (int j = 0; j < 4; ++j) {
      if (ks[j] >= T) {
        uint32_t pos = atomicAdd(&s_cnt, 1u);
        if (pos < 1024u)
          sel[pos] = ((unsigned long long)ks[j] << 32) |
                     (unsigned long long)(0xFFFFFFFFu - (bidx + j));
      }
    }
  }
  __syncthreads();

  // ---------- Bitonic sort, 1024 elems, descending ----------
  // Composite key = (valkey<<32)|(~idx): descending sort => value desc,
  // ties by ascending index (matches jax.lax.top_k). Taking the first 300
  // afterwards automatically keeps the lowest-index ties.
  for (int k = 2; k <= 1024; k <<= 1) {
    for (int j = k >> 1; j > 0; j >>= 1) {
      for (int i = tid; i < 1024; i += NTH) {
        int ixj = i ^ j;
        if (ixj > i) {
          unsigned long long a = sel[i], c = sel[ixj];
          bool desc = ((i & k) == 0);
          if ((a < c) == desc) { sel[i] = c; sel[ixj] = a; }
        }
      }
      __syncthreads();
    }
  }

  // ---------- Decode + emit ----------
  const float* boxrow = boxes + (size_t)b * QD * 4;
  const float img_h = (float)tsizes[b * 2 + 0];
  const float img_w = (float)tsizes[b * 2 + 1];
  float* out_scores = out;
  float* out_labels = out + (size_t)B * NSEL;
  float* out_boxes = out + 2 * (size_t)B * NSEL;
  for (int i = tid; i < NSEL; i += NTH) {
    unsigned long long ck = sel[i];
    uint32_t key = (uint32_t)(ck >> 32);
    uint32_t idx = 0xFFFFFFFFu - (uint32_t)(ck & 0xFFFFFFFFull);
    float lg = unfkey(key);
    float sc = 1.0f / (1.0f + expf(-lg));
    uint32_t q = idx / CD;
    uint32_t lab = idx - q * CD;
    size_t o = (size_t)b * NSEL + i;
    out_scores[o] = sc;
    out_labels[o] = (float)lab;
    float cx = boxrow[q * 4 + 0], cy = boxrow[q * 4 + 1];
    float w = boxrow[q * 4 + 2], h = boxrow[q * 4 + 3];
    out_boxes[o * 4 + 0] = (cx - 0.5f * w) * img_w;
    out_boxes[o * 4 + 1] = (cy - 0.5f * h) * img_h;
    out_boxes[o * 4 + 2] = (cx + 0.5f * w) * img_w;
    out_boxes[o * 4 + 3] = (cy + 0.5f * h) * img_h;
  }
}

extern "C" void kernel_launch(void* const* d_in, const int* in_sizes, int n_in,
                              void* d_out, int out_size, void* d_ws,
                              size_t ws_size, hipStream_t stream) {
  const float* logits = (const float*)d_in[0];
  const float* boxes = (const float*)d_in[1];
  const int* ts = (const int*)d_in[2];
  int B = in_sizes[0] / NQC;
  postprocess_topk_kernel<<<B, NTH, 0, stream>>>(logits, boxes, ts,
                                                 (float*)d_out, B);
}